// TopK_62397284876767
// MI455X (gfx1250) — compile-verified
//
#include <hip/hip_runtime.h>
#include <stdint.h>

// Top-k row masking: x [16,8,512,512] f32, keep top 128 per 512-row, zero rest.
// Bandwidth-bound target: 256 MiB @ 23.3 TB/s ~= 11 us.
// Design: ONE WAVE PER ROW (wave32), zero s_barrier usage. Row lives in 16
// VGPRs/lane; per-wave private 256-bin LDS histogram (in-order DS pipe +
// s_wait_dscnt is the only sync needed); cross-lane work via shfl/ballot.
// CDNA5 paths: global_load_async_to_lds_b128 / global_store_async_from_lds_b128
// (ASYNCcnt), ds atomics/b128, wave32 shuffles.

#define ROWLEN 512
#define KTOP   128
#define WPB    8            // waves (rows) per block
#define TPB    (32 * WPB)

__device__ __forceinline__ uint32_t fkey(float f) {
  // Order-preserving float -> u32 mapping (ascending).
  uint32_t u = __float_as_uint(f);
  return u ^ ((uint32_t)((int32_t)u >> 31) | 0x80000000u);
}

__global__ __launch_bounds__(TPB) void topk_mask_kernel(const float* __restrict__ x,
                                                        float* __restrict__ out) {
  __shared__ __align__(16) float    rowbuf[WPB][ROWLEN];  // 16 KB
  __shared__ __align__(16) uint32_t hist[WPB][256];       //  8 KB

  const int tid = threadIdx.x;
  const int w = tid >> 5;        // wave id in block = row slot
  const int l = tid & 31;        // lane
  const long long row = (long long)blockIdx.x * WPB + w;
  const float* src = x + row * ROWLEN;
  float* dst = out + row * ROWLEN;

  // LDS byte address of this lane's 64-byte slice (low 32 bits of generic ptr).
  const uint32_t ldsb = (uint32_t)(uintptr_t)(&rowbuf[w][l * 16]);

  // ---- Async global -> LDS: 64 B per lane (4 x b128), one asynccnt wait ----
  {
    uint64_t ga = (uint64_t)(uintptr_t)(src + l * 16);
    asm volatile(
        "global_load_async_to_lds_b128 %0, %1, off\n\t"
        "global_load_async_to_lds_b128 %0, %1, off offset:16\n\t"
        "global_load_async_to_lds_b128 %0, %1, off offset:32\n\t"
        "global_load_async_to_lds_b128 %0, %1, off offset:48\n\t"
        "s_wait_asynccnt 0"
        :: "v"(ldsb), "v"(ga) : "memory");
  }

  // ---- LDS -> registers: lane l owns columns [16l, 16l+16) ----
  float v[16];
  {
    const float4* p = (const float4*)(&rowbuf[w][l * 16]);
    float4 a = p[0], b = p[1], c = p[2], d = p[3];
    v[0]=a.x; v[1]=a.y; v[2]=a.z;  v[3]=a.w;
    v[4]=b.x; v[5]=b.y; v[6]=b.z;  v[7]=b.w;
    v[8]=c.x; v[9]=c.y; v[10]=c.z; v[11]=c.w;
    v[12]=d.x; v[13]=d.y; v[14]=d.z; v[15]=d.w;
  }
  uint32_t key[16];
  #pragma unroll
  for (int j = 0; j < 16; ++j) key[j] = fkey(v[j]);

  uint32_t* hw = hist[w];
  uint32_t prefix = 0u, krem = KTOP;
  const uint32_t himask[4] = {0x00000000u, 0xFF000000u, 0xFFFF0000u, 0xFFFFFF00u};

  // ---- 4-pass MSB-first radix select (wave-private, no barriers) ----
  #pragma unroll
  for (int pass = 0; pass < 4; ++pass) {
    const int shift = 24 - 8 * pass;
    const uint32_t hm = himask[pass];

    // Zero my 8 bins (bins [8l, 8l+8)); in-order DS pipe orders this vs atomics.
    {
      uint4 z = make_uint4(0u, 0u, 0u, 0u);
      *(uint4*)&hw[8 * l]     = z;
      *(uint4*)&hw[8 * l + 4] = z;
    }
    asm volatile("" ::: "memory");

    #pragma unroll
    for (int j = 0; j < 16; ++j) {
      if (((key[j] ^ prefix) & hm) == 0u)                 // candidate
        atomicAdd(&hw[(key[j] >> shift) & 255u], 1u);     // ds_add_u32
    }
    asm volatile("s_wait_dscnt 0" ::: "memory");          // atomics landed

    uint32_t h[8];
    {
      uint4 a = *(const uint4*)&hw[8 * l];
      uint4 b = *(const uint4*)&hw[8 * l + 4];
      h[0]=a.x; h[1]=a.y; h[2]=a.z; h[3]=a.w;
      h[4]=b.x; h[5]=b.y; h[6]=b.z; h[7]=b.w;
    }
    // In-lane suffix: tsuf[j] = count of my bins with index > j.
    uint32_t tsuf[8];
    tsuf[7] = 0u;
    #pragma unroll
    for (int j = 6; j >= 0; --j) tsuf[j] = tsuf[j + 1] + h[j + 1];
    uint32_t ctot = tsuf[0] + h[0];

    // Inclusive cross-lane prefix of per-lane totals (5 shfl_up steps).
    uint32_t P = ctot;
    #pragma unroll
    for (int off = 1; off < 32; off <<= 1) {
      uint32_t g = __shfl_up(P, off, 32);
      if (l >= off) P += g;
    }
    uint32_t grand = __shfl(P, 31, 32);
    uint32_t above = grand - P;            // candidates in bins owned by lanes > l

    // Exactly one (lane, j) satisfies: cnt_gt < krem <= cnt_gt + h[j].
    int fdig = -1; uint32_t fkrem = 0u;
    #pragma unroll
    for (int j = 0; j < 8; ++j) {
      uint32_t cgt = above + tsuf[j];
      if (cgt < krem && cgt + h[j] >= krem) { fdig = 8 * l + j; fkrem = krem - cgt; }
    }
    unsigned long long m = __ballot(fdig >= 0);
    int srcl = __ffsll(m) - 1;
    int dig  = __shfl(fdig, srcl, 32);
    krem     = (uint32_t)__shfl((int)fkrem, srcl, 32);
    prefix  |= ((uint32_t)dig) << shift;
  }

  const uint32_t T = prefix;               // key of the KTOP-th largest value
  // krem = number of ties (== T) to keep, lowest row index first.

  // ---- Tie ranks: lane prefix + in-lane running sum (index order) ----
  uint32_t e[16]; uint32_t ctie = 0u;
  #pragma unroll
  for (int j = 0; j < 16; ++j) { e[j] = (key[j] == T) ? 1u : 0u; ctie += e[j]; }
  uint32_t Pt = ctie;
  #pragma unroll
  for (int off = 1; off < 32; off <<= 1) {
    uint32_t g = __shfl_up(Pt, off, 32);
    if (l >= off) Pt += g;
  }
  uint32_t run = Pt - ctie;                // ties in lanes < l
  float o[16];
  #pragma unroll
  for (int j = 0; j < 16; ++j) {
    bool keep = (key[j] > T) || (e[j] && run < krem);
    run += e[j];
    o[j] = keep ? v[j] : 0.0f;
  }

  // ---- Registers -> LDS, then async LDS -> global (64 B per lane) ----
  {
    float4* p = (float4*)(&rowbuf[w][l * 16]);
    p[0] = make_float4(o[0],  o[1],  o[2],  o[3]);
    p[1] = make_float4(o[4],  o[5],  o[6],  o[7]);
    p[2] = make_float4(o[8],  o[9],  o[10], o[11]);
    p[3] = make_float4(o[12], o[13], o[14], o[15]);
  }
  asm volatile("s_wait_dscnt 0" ::: "memory");   // LDS writes visible to async engine
  {
    uint64_t ga = (uint64_t)(uintptr_t)(dst + l * 16);
    asm volatile(
        "global_store_async_from_lds_b128 %0, %1, off\n\t"
        "global_store_async_from_lds_b128 %0, %1, off offset:16\n\t"
        "global_store_async_from_lds_b128 %0, %1, off offset:32\n\t"
        "global_store_async_from_lds_b128 %0, %1, off offset:48\n\t"
        "s_wait_asynccnt 0"
        :: "v"(ga), "v"(ldsb) : "memory");
  }
}

extern "C" void kernel_launch(void* const* d_in, const int* in_sizes, int n_in,
                              void* d_out, int out_size, void* d_ws, size_t ws_size,
                              hipStream_t stream) {
  (void)n_in; (void)d_ws; (void)ws_size; (void)out_size;
  const float* x = (const float*)d_in[0];
  float* out = (float*)d_out;
  const int n = in_sizes[0];               // 16*8*512*512
  const int rows = n / ROWLEN;             // 65536
  const int blocks = rows / WPB;           // 8192
  topk_mask_kernel<<<dim3(blocks), dim3(TPB), 0, stream>>>(x, out);
}